// ShakespeareRNN_81046032876009
// MI455X (gfx1250) — compile-verified
//
#include <hip/hip_runtime.h>

// ---------------------------------------------------------------------------
// 2-layer GRU (HID=2048, batch 1) + decoder for MI455X (gfx1250, wave32).
//
// Memory-bound GEMV (~152MB fp32 weights, ~76 MFLOP; AI ~= 0.5 FLOP/B).
// HBM roofline ~6.5us first pass; weights are L2-resident (192MB) on replay.
//   * V_WMMA_F32_16X16X4_F32 tiles; vector replicated into all 16 B columns.
//   * Each wave owns TWO 16-row blocks (shared B from LDS, two independent
//     accumulator chains), 8-way K-split, unrolled 8x.
//   * Base pointers advanced once per 32-K group; inner offsets are
//     compile-time constants -> global_load_b64/ds_load_b64 with IOFFSET.
//   * fp32 partials in d_ws, deterministic tree-sum combines (no atomics).
// ---------------------------------------------------------------------------

typedef __attribute__((ext_vector_type(2))) float v2f;
typedef __attribute__((ext_vector_type(8))) float v8f;

#define HID    2048
#define EMB    6
#define OUT_N  100
#define GATES  (3 * HID)   // 6144
#define SPLITS 8           // waves per workgroup = K-splits per row block
#define KSTEP  32          // K per unrolled group (8 WMMA steps of 4)

// partial[row*SPLITS + wave] = dot(W[row, chunk], v[chunk])
// WMMA f32 16x16x4 layouts (ISA 7.12.2):
//   A (16x4): lane L -> M = L&15, half = L>>4; vgpr v holds K = 2*half + v
//   B (4x16): half selects K block {0,1}/{2,3}; vgpr v = K within block
//   D (16x16): lane L vgpr d -> (M = d + 8*half, N = L&15); cols identical.
__global__ __launch_bounds__(32 * SPLITS) void gemv32_wmma(
    const float* __restrict__ W, const float* __restrict__ vec,
    float* __restrict__ partial, int nrows, int K)
{
    __shared__ float sv[HID];
    {   // vectorized stage of the shared vector (K multiple of 4)
        const float4* v4 = (const float4*)vec;
        float4*       s4 = (float4*)sv;
        for (int i = threadIdx.x; i < (K >> 2); i += blockDim.x) s4[i] = v4[i];
    }
    __syncthreads();

    const int lane = threadIdx.x & 31;
    const int wv   = threadIdx.x >> 5;       // 0..SPLITS-1
    const int half = lane >> 4;              // K sub-block select
    const int m    = lane & 15;              // row within 16-row block
    const int rowBase = blockIdx.x * 32;     // two 16-row blocks per wave

    // Branch-free guard: clamp OOB rows onto the last valid row (writes are
    // guarded below), so the inner loop has no EXEC manipulation.
    const int rA = min(rowBase + m,      nrows - 1);
    const int rB = min(rowBase + 16 + m, nrows - 1);

    const int kPer = K / SPLITS;             // 256 for K=2048
    const int k0   = wv * kPer;
    const int koff = k0 + 2 * half;          // this lane's first K element

    // Loop-invariant bases; advanced once per KSTEP group so the unrolled
    // body uses constant instruction offsets.
    const float* p0 = W + (size_t)rA * K + koff;
    const float* p1 = W + (size_t)rB * K + koff;
    const float* pv = sv + koff;

    v8f acc0 = {0.f, 0.f, 0.f, 0.f, 0.f, 0.f, 0.f, 0.f};
    v8f acc1 = {0.f, 0.f, 0.f, 0.f, 0.f, 0.f, 0.f, 0.f};

    for (int kb = 0; kb < kPer; kb += KSTEP) {
        __builtin_prefetch(p0 + KSTEP, 0, 3);   // next 128B line, no branch
        __builtin_prefetch(p1 + KSTEP, 0, 3);
#pragma unroll
        for (int kk = 0; kk < KSTEP; kk += 4) {
            v2f a0, a1, b;
            __builtin_memcpy(&a0, p0 + kk, sizeof(v2f));  // global_load_b64
            __builtin_memcpy(&a1, p1 + kk, sizeof(v2f));  // global_load_b64
            __builtin_memcpy(&b,  pv + kk, sizeof(v2f));  // ds_load_b64
            // (neg_a, A, neg_b, B, c_mod, C, reuse_a, reuse_b)
            acc0 = __builtin_amdgcn_wmma_f32_16x16x4_f32(
                false, a0, false, b, (short)0, acc0, false, false);
            acc1 = __builtin_amdgcn_wmma_f32_16x16x4_f32(
                false, a1, false, b, (short)0, acc1, false, false);
        }
        p0 += KSTEP;
        p1 += KSTEP;
        pv += KSTEP;
    }

    if (m == 0) {                            // lanes 0 and 16 (N==0 columns)
#pragma unroll
        for (int d = 0; d < 8; ++d) {
            const int ra = rowBase + d + 8 * half;          // block 0
            if (ra < nrows) partial[(size_t)ra * SPLITS + wv] = acc0[d];
            const int rb = rowBase + 16 + d + 8 * half;     // block 1
            if (rb < nrows) partial[(size_t)rb * SPLITS + wv] = acc1[d];
        }
    }
}

__device__ __forceinline__ float sigmoidf_(float x) {
    return 1.0f / (1.0f + __expf(-x));
}

// Layer 0 combine: gi from tiny W_ih0 (EMB=6) inline; gh from partials.
__global__ void gru_combine_l0(
    const float* __restrict__ P,      // [GATES*SPLITS] partials of W_hh0 @ h0
    const int*   __restrict__ tok,
    const float* __restrict__ emb,    // [VOCAB, EMB]
    const float* __restrict__ W_ih,   // [GATES, EMB]
    const float* __restrict__ b_ih, const float* __restrict__ b_hh,
    const float* __restrict__ h_prev, float* __restrict__ h_out)
{
    const int j = blockIdx.x * blockDim.x + threadIdx.x;
    if (j >= HID) return;

    float x[EMB];
    const float* e = emb + (size_t)tok[0] * EMB;
#pragma unroll
    for (int q = 0; q < EMB; ++q) x[q] = e[q];

    float gi[3], gh[3];
#pragma unroll
    for (int g = 0; g < 3; ++g) {
        const int rr = g * HID + j;
        float s = b_ih[rr];
        const float* wr = W_ih + (size_t)rr * EMB;
#pragma unroll
        for (int q = 0; q < EMB; ++q) s += wr[q] * x[q];
        gi[g] = s;
        float t = b_hh[rr];
        const float* pp = P + (size_t)rr * SPLITS;
#pragma unroll
        for (int u = 0; u < SPLITS; ++u) t += pp[u];
        gh[g] = t;
    }
    const float r = sigmoidf_(gi[0] + gh[0]);
    const float z = sigmoidf_(gi[1] + gh[1]);
    const float n = tanhf(gi[2] + r * gh[2]);
    h_out[j] = (1.0f - z) * n + z * h_prev[j];
}

// Layer 1 combine: both gi and gh come from GEMV partials.
__global__ void gru_combine_l1(
    const float* __restrict__ Pi, const float* __restrict__ Ph,
    const float* __restrict__ b_ih, const float* __restrict__ b_hh,
    const float* __restrict__ h_prev, float* __restrict__ h_out)
{
    const int j = blockIdx.x * blockDim.x + threadIdx.x;
    if (j >= HID) return;

    float gi[3], gh[3];
#pragma unroll
    for (int g = 0; g < 3; ++g) {
        const int rr = g * HID + j;
        float s = b_ih[rr];
        const float* pi = Pi + (size_t)rr * SPLITS;
#pragma unroll
        for (int u = 0; u < SPLITS; ++u) s += pi[u];
        gi[g] = s;
        float t = b_hh[rr];
        const float* ph = Ph + (size_t)rr * SPLITS;
#pragma unroll
        for (int u = 0; u < SPLITS; ++u) t += ph[u];
        gh[g] = t;
    }
    const float r = sigmoidf_(gi[0] + gh[0]);
    const float z = sigmoidf_(gi[1] + gh[1]);
    const float n = tanhf(gi[2] + r * gh[2]);
    h_out[j] = (1.0f - z) * n + z * h_prev[j];
}

__global__ void dec_bias(const float* __restrict__ P,
                         const float* __restrict__ b,
                         float* __restrict__ out, int n)
{
    const int j = blockIdx.x * blockDim.x + threadIdx.x;
    if (j >= n) return;
    float s = b[j];
    const float* pp = P + (size_t)j * SPLITS;
#pragma unroll
    for (int u = 0; u < SPLITS; ++u) s += pp[u];
    out[j] = s;
}

extern "C" void kernel_launch(void* const* d_in, const int* in_sizes, int n_in,
                              void* d_out, int out_size, void* d_ws, size_t ws_size,
                              hipStream_t stream)
{
    const int*   tok    = (const int*)  d_in[0];
    const float* hidden = (const float*)d_in[1];   // [2,1,HID]
    const float* emb    = (const float*)d_in[2];
    const float* W_ih0  = (const float*)d_in[3];
    const float* W_hh0  = (const float*)d_in[4];
    const float* b_ih0  = (const float*)d_in[5];
    const float* b_hh0  = (const float*)d_in[6];
    const float* W_ih1  = (const float*)d_in[7];
    const float* W_hh1  = (const float*)d_in[8];
    const float* b_ih1  = (const float*)d_in[9];
    const float* b_hh1  = (const float*)d_in[10];
    const float* W_dec  = (const float*)d_in[11];
    const float* b_dec  = (const float*)d_in[12];

    float* out = (float*)d_out;          // [0..99]=logits, then h0, then h1
    float* h0  = out + OUT_N;            // new_hidden[0]
    float* h1  = out + OUT_N + HID;      // new_hidden[1]

    float* P0  = (float*)d_ws;                       // GATES*SPLITS
    float* P1i = P0  + (size_t)GATES * SPLITS;       // GATES*SPLITS
    float* P1h = P1i + (size_t)GATES * SPLITS;       // GATES*SPLITS
    float* Pd  = P1h + (size_t)GATES * SPLITS;       // OUT_N*SPLITS

    const dim3 blk(32 * SPLITS);

    // Layer 0: gh0 = W_hh0 @ hidden[0]
    gemv32_wmma<<<GATES / 32, blk, 0, stream>>>(W_hh0, hidden, P0, GATES, HID);
    gru_combine_l0<<<HID / 256, dim3(256), 0, stream>>>(
        P0, tok, emb, W_ih0, b_ih0, b_hh0, hidden, h0);

    // Layer 1: gi1 = W_ih1 @ h0 ; gh1 = W_hh1 @ hidden[1]
    gemv32_wmma<<<GATES / 32, blk, 0, stream>>>(W_ih1, h0,           P1i, GATES, HID);
    gemv32_wmma<<<GATES / 32, blk, 0, stream>>>(W_hh1, hidden + HID, P1h, GATES, HID);
    gru_combine_l1<<<HID / 256, dim3(256), 0, stream>>>(
        P1i, P1h, b_ih1, b_hh1, hidden + HID, h1);

    // Decoder: logits = W_dec @ h1 + b_dec   (rows clamped, stores guarded)
    gemv32_wmma<<<(OUT_N + 31) / 32, blk, 0, stream>>>(W_dec, h1, Pd, OUT_N, HID);
    dec_bias<<<1, dim3(128), 0, stream>>>(Pd, b_dec, out, OUT_N);
}